// QueryAndGroup_71829033058389
// MI455X (gfx1250) — compile-verified
//
#include <hip/hip_runtime.h>
#include <hip/hip_bf16.h>

typedef __attribute__((ext_vector_type(2))) float v2f;
typedef __attribute__((ext_vector_type(8))) float v8f;

#define NB      4        // batches
#define NPTS    16384    // cloud points per batch
#define NQ      2048     // query points per batch
#define NS      32       // samples per query
#define NC      64       // feature channels
#define R2      0.01f    // radius^2
#define OUTW    67       // 3 + 64

// ---------------------------------------------------------------------------
// Kernel 0: pack points as [x, y, z, |p|^2] so the WMMA B-operand is one b64
// load per lane. (lanes 0-15 -> (K0,K1)=(x,y); lanes 16-31 -> (K2,K3)=(z,n2))
// ---------------------------------------------------------------------------
__global__ __launch_bounds__(256) void pack_kernel(
    const float* __restrict__ xyz, float4* __restrict__ pack) {
  int p = blockIdx.x * 256 + threadIdx.x;
  if (p < NB * NPTS) {
    float x = xyz[3 * p + 0];
    float y = xyz[3 * p + 1];
    float z = xyz[3 * p + 2];
    float4 v;
    v.x = x; v.y = y; v.z = z; v.w = x * x + y * y + z * z;
    pack[p] = v;
  }
}

// ---------------------------------------------------------------------------
// Kernel 1: ball query via V_WMMA_F32_16X16X4_F32.
//   A (16x4)  row m = (-2qx, -2qy, -2qz, 1)      -> per-wave constant
//   B (4x16)  col n = ( px,   py,   pz, |p|^2)   -> one b64 load per lane
//   D = |p|^2 - 2 q.p ; hit iff D < r^2 - |q|^2  (per-row threshold)
// One wave handles 16 queries; software-pipelined x2 (two B tiles in flight);
// empty-mask tiles (the common case) skip all select/append work via a
// scalar branch on the OR of the 8 uniform cmp masks.
// ---------------------------------------------------------------------------
__global__ __launch_bounds__(256) void ballquery_wmma_kernel(
    const float* __restrict__ new_xyz, const float4* __restrict__ pack,
    int* __restrict__ idx_out) {
  __shared__ float qn2s[8 * 16];       // per-wave |q|^2
  __shared__ int   idxs[8 * 16 * NS];  // per-wave index lists (16 KB)

  const int lane = threadIdx.x & 31;
  const int wave = threadIdx.x >> 5;
  const int tile = blockIdx.x * 8 + wave;   // 0..511
  const int b    = tile >> 7;               // tile / 128
  const int s0   = (tile & 127) << 4;       // 16 queries per tile
  const int m    = lane & 15;
  const bool lo  = (lane < 16);

  // Load this lane's query point (query m of the tile).
  const float* q = new_xyz + ((size_t)b * NQ + s0 + m) * 3;
  const float qx = q[0], qy = q[1], qz = q[2];
  const float qn2 = qx * qx + qy * qy + qz * qz;
  if (lo) qn2s[wave * 16 + m] = qn2;
  __syncthreads();

  // A operand per ISA layout: lanes 0-15 hold (K0,K1), lanes 16-31 (K2,K3).
  v2f A;
  A.x = lo ? (-2.0f * qx) : (-2.0f * qz);
  A.y = lo ? (-2.0f * qy) : 1.0f;

  // Per-row thresholds for the C/D layout (VGPR r: rows r / r+8).
  const int hi8 = lo ? 0 : 8;
  const float* qb = qn2s + wave * 16 + hi8;
  const float t0 = R2 - qb[0], t1 = R2 - qb[1], t2 = R2 - qb[2], t3 = R2 - qb[3];
  const float t4 = R2 - qb[4], t5 = R2 - qb[5], t6 = R2 - qb[6], t7 = R2 - qb[7];

  // Per-lane base address into packed points for the B operand.
  const char* pbase = (const char*)(pack + (size_t)b * NPTS) + m * 16 + (lo ? 0 : 8);

  int  cnt  = lo ? 0 : NS;  // hi lanes always report "done"
  int* myid = idxs + (wave * 16 + m) * NS;

  // Process one 16-candidate tile; returns true when every query is done.
  auto process = [&](v2f Bv, int n0) -> bool {
    v8f c = {};
    v8f d = __builtin_amdgcn_wmma_f32_16x16x4_f32(
        /*neg_a=*/false, A, /*neg_b=*/false, Bv,
        /*c_mod=*/(short)0, c, /*reuse_a=*/false, /*reuse_b=*/false);

    // Ballot per accumulator VGPR: bits 0-15 = row r, bits 16-31 = row r+8.
    unsigned mk0 = __builtin_amdgcn_ballot_w32(d[0] < t0);
    unsigned mk1 = __builtin_amdgcn_ballot_w32(d[1] < t1);
    unsigned mk2 = __builtin_amdgcn_ballot_w32(d[2] < t2);
    unsigned mk3 = __builtin_amdgcn_ballot_w32(d[3] < t3);
    unsigned mk4 = __builtin_amdgcn_ballot_w32(d[4] < t4);
    unsigned mk5 = __builtin_amdgcn_ballot_w32(d[5] < t5);
    unsigned mk6 = __builtin_amdgcn_ballot_w32(d[6] < t6);
    unsigned mk7 = __builtin_amdgcn_ballot_w32(d[7] < t7);

    // Uniform (scalar) skip of the empty tile -- the common case (~99%).
    unsigned any = mk0 | mk1 | mk2 | mk3 | mk4 | mk5 | mk6 | mk7;
    if (any) {
      // Lane L < 16 owns query row L: select its 16-bit column mask.
      const int r = lane & 7;
      unsigned mm = mk0;
      mm = (r == 1) ? mk1 : mm;
      mm = (r == 2) ? mk2 : mm;
      mm = (r == 3) ? mk3 : mm;
      mm = (r == 4) ? mk4 : mm;
      mm = (r == 5) ? mk5 : mm;
      mm = (r == 6) ? mk6 : mm;
      mm = (r == 7) ? mk7 : mm;
      unsigned mym = lo ? ((lane < 8) ? (mm & 0xFFFFu) : (mm >> 16)) : 0u;

      // Append hit indices in ascending order (matches reference top_k).
      while (mym && cnt < NS) {
        int kbit = __builtin_ctz(mym);
        myid[cnt++] = n0 + kbit;
        mym &= mym - 1;
      }

      // cnt only changes on hits, so the done-check lives here.
      if (__builtin_amdgcn_ballot_w32(cnt >= NS) == 0xFFFFFFFFu) return true;
    }
    return false;
  };

  // Software pipeline: keep two B tiles (32 candidates) in flight.
  v2f B0 = *(const v2f*)(pbase + 0);
  v2f B1 = *(const v2f*)(pbase + 16 * 16);
  for (int n0 = 0; n0 < NPTS; n0 += 32) {
    const int np = n0 + 32;
    const size_t o2 = (np      < NPTS) ? (size_t)np * 16        : 0;
    const size_t o3 = (np + 16 < NPTS) ? (size_t)(np + 16) * 16 : 0;
    v2f B2 = *(const v2f*)(pbase + o2);   // issued before B0/B1 consumed
    v2f B3 = *(const v2f*)(pbase + o3);
    if (process(B0, n0)) break;
    if (process(B1, n0 + 16)) break;
    B0 = B2;
    B1 = B3;
  }

  // Tail: fill remainder with first hit; no hits -> N-1 (reference clip).
  if (lo) {
    int first = (cnt > 0) ? myid[0] : (NPTS - 1);
    for (int k = cnt; k < NS; ++k) myid[k] = first;
  }

  // Write the wave's 16x32 index block, coalesced.
  int* gout = idx_out + ((size_t)b * NQ + s0) * NS;
  for (int row = 0; row < 16; ++row) {
    gout[row * NS + lane] = idxs[(wave * 16 + row) * NS + lane];
  }
}

// ---------------------------------------------------------------------------
// Kernel 2: gather + concat. One block per query; threads stride the 32x67
// output row. Features (17 MB) are L2-resident on MI455X (192 MB L2).
// ---------------------------------------------------------------------------
__global__ __launch_bounds__(256) void gather_kernel(
    const float* __restrict__ xyz, const float* __restrict__ new_xyz,
    const float* __restrict__ feat, const int* __restrict__ idx,
    float* __restrict__ out) {
  const int bs = blockIdx.x;        // b*NQ + s
  const int b  = bs >> 11;          // / 2048
  const int* myidx = idx + (size_t)bs * NS;
  float* o = out + (size_t)bs * NS * OUTW;

  for (int e = threadIdx.x; e < NS * OUTW; e += 256) {
    int k = e / OUTW;
    int c = e - k * OUTW;
    int id = myidx[k];
    float v;
    if (c < 3) {
      v = xyz[((size_t)b * NPTS + id) * 3 + c] - new_xyz[(size_t)bs * 3 + c];
    } else {
      v = feat[((size_t)b * NPTS + id) * NC + (c - 3)];
    }
    o[e] = v;
  }
}

extern "C" void kernel_launch(void* const* d_in, const int* in_sizes, int n_in,
                              void* d_out, int out_size, void* d_ws, size_t ws_size,
                              hipStream_t stream) {
  const float* xyz      = (const float*)d_in[0];  // (4, 16384, 3)
  const float* new_xyz  = (const float*)d_in[1];  // (4, 2048, 3)
  const float* features = (const float*)d_in[2];  // (4, 16384, 64)
  float* out = (float*)d_out;                     // (4, 2048, 32, 67)

  // Workspace layout: pack (1 MB) | idx (1 MB)
  float4* pack = (float4*)d_ws;
  int*    idx  = (int*)((char*)d_ws + (size_t)NB * NPTS * sizeof(float4));

  pack_kernel<<<(NB * NPTS + 255) / 256, 256, 0, stream>>>(xyz, pack);
  ballquery_wmma_kernel<<<64, 256, 0, stream>>>(new_xyz, pack, idx);
  gather_kernel<<<NB * NQ, 256, 0, stream>>>(xyz, new_xyz, features, idx, out);
}